// VarianceFilter2D_68745246540293
// MI455X (gfx1250) — compile-verified
//
#include <hip/hip_runtime.h>

typedef __attribute__((ext_vector_type(2))) float v2f;
typedef __attribute__((ext_vector_type(8))) float v8f;

#define IMG_N 1024
#define PAD   3
#define WSZ   7
#define XSTR  23   // 22-col padded tile row stride (odd -> conflict-free column walks)
#define HSTR  17   // H row stride (lane groups land in disjoint bank ranges)

__device__ __forceinline__ int reflect_idx(int p) {
    // jnp.pad mode='reflect' (edge not repeated); pad=3 < N so one reflection suffices
    p = (p < 0) ? -p : p;
    p = (p > IMG_N - 1) ? (2 * (IMG_N - 1) - p) : p;
    return p;
}

__global__ __launch_bounds__(256)
void VarianceFilter2D_kernel(const float* __restrict__ img, float* __restrict__ out) {
    const int lane = threadIdx.x & 31;
    const int wid  = threadIdx.x >> 5;          // 8 waves / block, one 16x16 tile each
    const int c0   = blockIdx.x * 128 + wid * 16;
    const int r0   = blockIdx.y * 16;
    const long ch_off = (long)blockIdx.z * ((long)IMG_N * IMG_N);

    __shared__ float s_xt[8][22 * XSTR];        // padded input tile  (per wave)
    __shared__ float s_h1[8][24 * HSTR];        // horizontal mean of x
    __shared__ float s_h2[8][24 * HSTR];        // horizontal mean of x^2
    float* xt = s_xt[wid];
    float* h1 = s_h1[wid];
    float* h2 = s_h2[wid];

    // ---- Phase A: coalesced load of 22x22 reflect-padded tile into LDS ----
    if (lane < 22) {
        const int cg = reflect_idx(c0 - PAD + lane);
        #pragma unroll 4
        for (int r = 0; r < 22; ++r) {
            const int rg = reflect_idx(r0 - PAD + r);
            xt[r * XSTR + lane] = img[ch_off + (long)rg * IMG_N + cg];
        }
    }
    __syncthreads();

    // ---- Phase B: horizontal 7-tap sliding box means (one row per lane) ----
    const float inv7 = 1.0f / 7.0f;
    if (lane < 22) {
        const int r = lane;
        float s = 0.f, s2 = 0.f;
        for (int j = 0; j < 22; ++j) {
            const float v = xt[r * XSTR + j];
            s += v; s2 += v * v;
            if (j >= WSZ) {
                const float u = xt[r * XSTR + j - WSZ];
                s -= u; s2 -= u * u;
            }
            if (j >= WSZ - 1) {
                h1[r * HSTR + (j - (WSZ - 1))] = s  * inv7;
                h2[r * HSTR + (j - (WSZ - 1))] = s2 * inv7;
            }
        }
    } else if (lane < 24) {
        // zero K-padding rows 22..23 so the WMMA K loop can run to 24
        #pragma unroll
        for (int t = 0; t < 16; ++t) {
            h1[lane * HSTR + t] = 0.f;
            h2[lane * HSTR + t] = 0.f;
        }
    }
    __syncthreads();

    // ---- Phase C: vertical 7-tap box mean as chained V_WMMA_F32_16X16X4_F32 ----
    // D(16x16) = Wv(16x24) x H(24x16), Wv[m][k] = (m <= k <= m+6) / 7
    const int m  = lane & 15;                   // A-matrix row / D column index
    const int hi = lane >> 4;                   // lanes 16..31 carry K+2,K+3 halves
    v8f acc1 = {};                              // mean accumulator
    v8f acc2 = {};                              // mean-of-squares accumulator
    #pragma unroll
    for (int kk = 0; kk < 6; ++kk) {
        const int kb = 4 * kk + (hi ? 2 : 0);
        // A operand: band-matrix slice, computed per lane (ISA 16x4 f32 A layout)
        v2f a;
        a.x = (kb     >= m && kb     <= m + 6) ? inv7 : 0.f;
        a.y = (kb + 1 >= m && kb + 1 <= m + 6) ? inv7 : 0.f;
        // B operand: rows kb, kb+1 of H, column = lane%16 (mirrored lane layout)
        v2f b1, b2;
        b1.x = h1[kb * HSTR + m];  b1.y = h1[(kb + 1) * HSTR + m];
        b2.x = h2[kb * HSTR + m];  b2.y = h2[(kb + 1) * HSTR + m];
        acc1 = __builtin_amdgcn_wmma_f32_16x16x4_f32(
                   false, a, false, b1, (short)0, acc1, false, false);
        acc2 = __builtin_amdgcn_wmma_f32_16x16x4_f32(
                   false, a, false, b2, (short)0, acc2, false, false);
    }

    // ---- Phase D: variance = E[x^2] - E[x]^2, store from D-matrix layout ----
    // VGPR v holds row (v)      for lanes 0..15  (col = lane)
    //              row (v + 8)  for lanes 16..31 (col = lane-16)
    #pragma unroll
    for (int v = 0; v < 8; ++v) {
        const int row = r0 + v + (hi ? 8 : 0);
        const float m1  = acc1[v];
        const float var = acc2[v] - m1 * m1;
        out[ch_off + (long)row * IMG_N + (c0 + m)] = var;
    }
}

extern "C" void kernel_launch(void* const* d_in, const int* in_sizes, int n_in,
                              void* d_out, int out_size, void* d_ws, size_t ws_size,
                              hipStream_t stream) {
    (void)in_sizes; (void)n_in; (void)out_size; (void)d_ws; (void)ws_size;
    const float* img = (const float*)d_in[0];   // (64, 1024, 1024) f32
    // d_in[1] = window_size (fixed at 7 by the reference; compiled in)
    float* out = (float*)d_out;                 // (64, 1, 1024, 1024) f32, same flat layout

    dim3 grid(IMG_N / 128, IMG_N / 16, 64);     // 8 x 64 x 64 blocks
    dim3 block(256);                            // 8 wave32 waves, one 16x16 tile each
    VarianceFilter2D_kernel<<<grid, block, 0, stream>>>(img, out);
}